// MeanAggregator_63814624084657
// MI455X (gfx1250) — compile-verified
//
#include <hip/hip_runtime.h>
#include <hip/hip_bf16.h>
#include <stdint.h>

// MeanAggregator on MI455X (gfx1250):
//   out[i,:] = (1/N) * sum_{s<16} nodes[edges[i,s], :]
//
// Memory-bound gather-reduce (AI ~0.25 FLOP/byte); nodes table (20.5MB) is
// L2-resident (192MB L2). Per wave: one node at a time, 16 neighbor rows
// (512B each) gathered into LDS with GLOBAL_LOAD_ASYNC_TO_LDS_B128 (GVS form:
// scalar row base + invariant lane*16 offset), double-buffered, overlapped
// with the previous node's LDS reduction via split s_wait_asynccnt.
// Edge indices for a node are fetched up-front as 4x int4 so the async issue
// burst is not serialized on per-index load latency.

#define D_FEAT 128
#define DEG 16
#define WAVES_PER_BLOCK 4
#define THREADS (WAVES_PER_BLOCK * 32)
#define ROW_BYTES (D_FEAT * 4)        /* 512 B per feature row  */
#define BUF_FLOATS (DEG * D_FEAT)     /* 2048 floats = 8 KB     */

__device__ __forceinline__ void issue_node_gather(const float* __restrict__ nodes,
                                                  const int* __restrict__ edges,
                                                  int nd_uniform,
                                                  unsigned lds_base_lane,
                                                  unsigned voff) {
    // One 64B row of indices: 4 independent b128 loads, single wait.
    const int4* erow = (const int4*)(edges + (size_t)(unsigned)nd_uniform * DEG);
    int4 e0 = erow[0];
    int4 e1 = erow[1];
    int4 e2 = erow[2];
    int4 e3 = erow[3];
    int idx[DEG] = { e0.x, e0.y, e0.z, e0.w,
                     e1.x, e1.y, e1.z, e1.w,
                     e2.x, e2.y, e2.z, e2.w,
                     e3.x, e3.y, e3.z, e3.w };
#pragma unroll
    for (int s = 0; s < DEG; ++s) {
        // Index is wave-uniform: hoist to SGPR so the row base is scalar math
        // and the async load can use the saddr (GVS) form.
        int u = __builtin_amdgcn_readfirstlane(idx[s]);
        const char* base = (const char*)nodes + (size_t)(unsigned)u * ROW_BYTES;
        unsigned ldst = lds_base_lane + (unsigned)(s * ROW_BYTES);
        // ASYNCcnt-tracked gather:
        //   each lane copies 16 bytes from global address base + voff to the LDS address in ldst
        asm volatile("global_load_async_to_lds_b128 %0, %1, %2"
                     :: "v"(ldst), "v"(voff), "s"(base)
                     : "memory");
    }
}

__global__ __launch_bounds__(THREADS) void
mean_agg_async_kernel(const float* __restrict__ nodes,
                      const int* __restrict__ edges,
                      float* __restrict__ out,
                      int n_nodes, int wave_stride, float scale) {
    // [wave][2 buffers][DEG rows][D_FEAT floats] = 4*2*8KB = 64KB
    __shared__ __align__(16) float lds[WAVES_PER_BLOCK * 2 * BUF_FLOATS];

    const int lane = (int)(threadIdx.x & 31u);
    const int wave = (int)(threadIdx.x >> 5);
    // Wave-uniform by construction; make it explicit for the scalar unit.
    const int gwave = __builtin_amdgcn_readfirstlane(
        (int)blockIdx.x * WAVES_PER_BLOCK + wave);

    float* buf0 = &lds[(wave * 2 + 0) * BUF_FLOATS];
    float* buf1 = &lds[(wave * 2 + 1) * BUF_FLOATS];
    // Low 32 bits of a generic LDS pointer == workgroup-relative LDS byte
    // offset (flat->LDS aperture truncation) == what the async VDST expects.
    const unsigned voff = (unsigned)lane * 16u;           // global sub-row offset
    const unsigned lb0 = (unsigned)(uintptr_t)buf0 + voff; // LDS dst, buffer 0
    const unsigned lb1 = (unsigned)(uintptr_t)buf1 + voff; // LDS dst, buffer 1

    int nd = gwave;
    if (nd >= n_nodes) return;

    // Prologue: fill buffer 0 for the first node.
    issue_node_gather(nodes, edges, nd, lb0, voff);

    int cur = 0;
    while (nd < n_nodes) {
        const int nxt = nd + wave_stride;
        const bool more = (nxt < n_nodes);

        // Prefetch next node's 8KB into the other buffer.
        if (more)
            issue_node_gather(nodes, edges, nxt, cur ? lb0 : lb1, voff);

        // Async loads complete in order: with 32 outstanding, <=16 means the
        // current buffer's 16 are done; on the tail, drain fully.
        if (more)
            asm volatile("s_wait_asynccnt 16" ::: "memory");
        else
            asm volatile("s_wait_asynccnt 0" ::: "memory");

        // Reduce DEG rows from LDS; lane owns features [4*lane, 4*lane+3].
        const float* buf = cur ? buf1 : buf0;
        float4 acc = make_float4(0.f, 0.f, 0.f, 0.f);
#pragma unroll
        for (int s = 0; s < DEG; ++s) {
            float4 v = *(const float4*)&buf[s * D_FEAT + lane * 4];
            acc.x += v.x; acc.y += v.y; acc.z += v.z; acc.w += v.w;
        }
        float4 r = make_float4(acc.x * scale, acc.y * scale,
                               acc.z * scale, acc.w * scale);
        *(float4*)&out[(size_t)nd * D_FEAT + lane * 4] = r;

        nd = nxt;
        cur ^= 1;
    }
}

extern "C" void kernel_launch(void* const* d_in, const int* in_sizes, int n_in,
                              void* d_out, int out_size, void* d_ws, size_t ws_size,
                              hipStream_t stream) {
    const float* nodes = (const float*)d_in[0];
    const int*   edges = (const int*)d_in[1];
    float*       out   = (float*)d_out;

    const int n_nodes = in_sizes[0] / D_FEAT;           // 40000
    const float scale = 1.0f / (float)n_nodes;          // reference divides by node_count

    const int blocks = 2500;                            // 10000 waves -> 4 nodes/wave
    const int wave_stride = blocks * WAVES_PER_BLOCK;

    mean_agg_async_kernel<<<blocks, THREADS, 0, stream>>>(
        nodes, edges, out, n_nodes, wave_stride, scale);
}